// MSCALite_4174708211848
// MI455X (gfx1250) — compile-verified
//
#include <hip/hip_runtime.h>
#include <hip/hip_bf16.h>
#include <math.h>

#define C_CH 128
#define H_DIM 128
#define W_DIM 128
#define B_DIM 16
#define HW (H_DIM * W_DIM)

typedef float v2f __attribute__((ext_vector_type(2)));
typedef float v8f __attribute__((ext_vector_type(8)));
typedef int   v4i __attribute__((ext_vector_type(4)));

#if defined(__has_builtin)
#if __has_builtin(__builtin_amdgcn_global_load_async_to_lds_b128)
#define HAVE_ASYNC_LDS 1
#endif
#if __has_builtin(__builtin_amdgcn_s_wait_asynccnt)
#define HAVE_WAIT_ASYNC 1
#endif
#endif

// ---------------------------------------------------------------------------
// Kernel 1: depthwise 5x5, x -> attn.  One (b,c) plane per blockIdx.z,
// 16x16 output tile per block, 20x20 input tile staged in LDS.
// ---------------------------------------------------------------------------
__global__ __launch_bounds__(256)
void msca_dw5x5(const float* __restrict__ x, const float* __restrict__ w,
                float* __restrict__ attn) {
    __shared__ float tile[20][20];
    __shared__ float wk[25];

    const int p   = blockIdx.z;            // b*C + c
    const int c   = p & (C_CH - 1);
    const int tx0 = blockIdx.x * 16;
    const int ty0 = blockIdx.y * 16;
    const float* xp = x + (size_t)p * HW;

    if (threadIdx.x < 25) wk[threadIdx.x] = w[c * 25 + threadIdx.x];

    for (int i = threadIdx.x; i < 20 * 20; i += 256) {
        int r  = i / 20, q = i % 20;
        int gy = ty0 + r - 2, gx = tx0 + q - 2;
        float v = 0.0f;
        if (gy >= 0 && gy < H_DIM && gx >= 0 && gx < W_DIM) v = xp[gy * W_DIM + gx];
        tile[r][q] = v;
    }
    __syncthreads();

    const int ty = threadIdx.x >> 4;
    const int tx = threadIdx.x & 15;
    float acc = 0.0f;
#pragma unroll
    for (int ky = 0; ky < 5; ++ky)
#pragma unroll
        for (int kx = 0; kx < 5; ++kx)
            acc += tile[ty + ky][tx + kx] * wk[ky * 5 + kx];

    attn[(size_t)p * HW + (size_t)(ty0 + ty) * W_DIM + (tx0 + tx)] = acc;
}

// ---------------------------------------------------------------------------
// Kernel 2: fused separable branches, attn -> s = s1+s2+s3.
// 32x32 output tile, 52x52 input tile (halo 10) in LDS.
// Horizontal 7/11/21-tap convs into LDS, then vertical convs summed.
// ---------------------------------------------------------------------------
__global__ __launch_bounds__(256)
void msca_sep(const float* __restrict__ attn,
              const float* __restrict__ w1h, const float* __restrict__ w1v,
              const float* __restrict__ w2h, const float* __restrict__ w2v,
              const float* __restrict__ w3h, const float* __restrict__ w3v,
              float* __restrict__ s) {
    __shared__ float tile[52][52];          // 10816 B
    __shared__ float h1[52][32];            // 6656 B
    __shared__ float h2[52][32];
    __shared__ float h3[52][32];
    __shared__ float wh1[7], wv1[7], wh2[11], wv2[11], wh3[21], wv3[21];

    const int p   = blockIdx.z;
    const int c   = p & (C_CH - 1);
    const int tx0 = blockIdx.x * 32;
    const int ty0 = blockIdx.y * 32;
    const float* ap = attn + (size_t)p * HW;

    if (threadIdx.x < 7)  { wh1[threadIdx.x] = w1h[c * 7  + threadIdx.x];
                            wv1[threadIdx.x] = w1v[c * 7  + threadIdx.x]; }
    else if (threadIdx.x >= 32 && threadIdx.x < 43) {
        int i = threadIdx.x - 32;
        wh2[i] = w2h[c * 11 + i]; wv2[i] = w2v[c * 11 + i];
    } else if (threadIdx.x >= 64 && threadIdx.x < 85) {
        int i = threadIdx.x - 64;
        wh3[i] = w3h[c * 21 + i]; wv3[i] = w3v[c * 21 + i];
    }

    for (int i = threadIdx.x; i < 52 * 52; i += 256) {
        int r  = i / 52, q = i % 52;
        int gy = ty0 + r - 10, gx = tx0 + q - 10;
        float v = 0.0f;
        if (gy >= 0 && gy < H_DIM && gx >= 0 && gx < W_DIM) v = ap[gy * W_DIM + gx];
        tile[r][q] = v;
    }
    __syncthreads();

    // horizontal passes over all 52 rows x 32 output cols
    for (int i = threadIdx.x; i < 52 * 32; i += 256) {
        int r = i >> 5, j = i & 31;
        float a1 = 0.0f, a2 = 0.0f, a3 = 0.0f;
#pragma unroll
        for (int d = 0; d < 7; ++d)  a1 += tile[r][j + 7 + d] * wh1[d];
#pragma unroll
        for (int d = 0; d < 11; ++d) a2 += tile[r][j + 5 + d] * wh2[d];
#pragma unroll
        for (int d = 0; d < 21; ++d) a3 += tile[r][j + d]     * wh3[d];
        h1[r][j] = a1; h2[r][j] = a2; h3[r][j] = a3;
    }
    __syncthreads();

    // vertical passes, summed
    float* sp = s + (size_t)p * HW;
    for (int i = threadIdx.x; i < 32 * 32; i += 256) {
        int r = i >> 5, j = i & 31;
        float acc = 0.0f;
#pragma unroll
        for (int d = 0; d < 7; ++d)  acc += h1[r + 7 + d][j] * wv1[d];
#pragma unroll
        for (int d = 0; d < 11; ++d) acc += h2[r + 5 + d][j] * wv2[d];
#pragma unroll
        for (int d = 0; d < 21; ++d) acc += h3[r + d][j]     * wv3[d];
        sp[(size_t)(ty0 + r) * W_DIM + (tx0 + j)] = acc;
    }
}

// ---------------------------------------------------------------------------
// Kernel 3: 1x1 projection (WMMA f32 16x16x4, K=128) + sigmoid gate.
// Block = 8 waves; wave w owns M-rows [16w, 16w+16) of w_proj, block owns a
// 64-pixel column slab of one batch image. s-slab staged in LDS (32 KB),
// via async global->LDS (b128, ASYNCcnt) when the toolchain exposes it.
// ---------------------------------------------------------------------------
__global__ __launch_bounds__(256)
void msca_proj_gate(const float* __restrict__ s, const float* __restrict__ wproj,
                    const float* __restrict__ x, float* __restrict__ out) {
    __shared__ float smem[C_CH * 64];       // 32 KB: s[ch][col]

    const int b   = blockIdx.y;
    const int n0b = blockIdx.x * 64;
    const size_t plane = (size_t)b * C_CH * HW;
    const float* sb = s + plane;

#if defined(HAVE_ASYNC_LDS)
    // 2048 b128 transfers, 8 per thread; both sides 16B-aligned.
    {
        typedef __attribute__((address_space(1))) v4i* gv4i_p;
        typedef __attribute__((address_space(3))) v4i* lv4i_p;
#pragma unroll
        for (int j = 0; j < 8; ++j) {
            int i    = threadIdx.x + 256 * j;        // b128 chunk index
            int row  = i >> 4;                       // 16 chunks per 64-float row
            int col4 = (i & 15) << 2;                // float offset within row
            const float* g = sb + (size_t)row * HW + n0b + col4;
            float* l = &smem[row * 64 + col4];
            __builtin_amdgcn_global_load_async_to_lds_b128(
                (gv4i_p)(void*)g, (lv4i_p)(void*)l, 0, 0);
        }
    }
#if defined(HAVE_WAIT_ASYNC)
    __builtin_amdgcn_s_wait_asynccnt(0);
#else
    asm volatile("s_wait_asynccnt 0x0" ::: "memory");
#endif
    __syncthreads();
#else
    for (int i = threadIdx.x; i < C_CH * 64; i += 256) {
        int row = i >> 6, col = i & 63;
        smem[i] = sb[(size_t)row * HW + n0b + col];
    }
    __syncthreads();
#endif

    const int wave = threadIdx.x >> 5;
    const int lane = threadIdx.x & 31;
    const int m0   = wave * 16;
    const int mrow = m0 + (lane & 15);          // A: lanes 0..15 carry M, hi half K+2
    const int koff = (lane < 16) ? 0 : 2;

    // Preload all A fragments (w_proj rows m0..m0+15, full K=128): 32 x v2f
    v2f afrag[32];
#pragma unroll
    for (int ks = 0; ks < 32; ++ks) {
        int kb = 4 * ks + koff;
        afrag[ks].x = wproj[mrow * C_CH + kb];
        afrag[ks].y = wproj[mrow * C_CH + kb + 1];
    }

    const float* xb = x + plane;
    float* ob = out + plane;
    const int mhalf = (lane < 16) ? 0 : 8;      // C/D: VGPR r -> M=r / M=r+8

    for (int nt = 0; nt < 4; ++nt) {
        const int ncol = nt * 16 + (lane & 15); // B/D column within slab
        v8f acc = {};
#pragma unroll
        for (int ks = 0; ks < 32; ++ks) {
            int kb = 4 * ks + koff;             // B: lanes 0..15 rows K,K+1; hi half K+2,K+3
            v2f bfrag;
            bfrag.x = smem[kb * 64 + ncol];
            bfrag.y = smem[(kb + 1) * 64 + ncol];
            acc = __builtin_amdgcn_wmma_f32_16x16x4_f32(
                false, afrag[ks], false, bfrag, (short)0, acc, false, false);
        }
        const int n = n0b + ncol;
#pragma unroll
        for (int r = 0; r < 8; ++r) {
            int m = m0 + mhalf + r;
            size_t off = (size_t)m * HW + n;
            float g = 1.0f / (1.0f + __expf(-acc[r]));
            ob[off] = xb[off] * g;
        }
    }
}

// ---------------------------------------------------------------------------
extern "C" void kernel_launch(void* const* d_in, const int* in_sizes, int n_in,
                              void* d_out, int out_size, void* d_ws, size_t ws_size,
                              hipStream_t stream) {
    const float* x      = (const float*)d_in[0];
    const float* w_dw5  = (const float*)d_in[1];
    const float* w_s1h  = (const float*)d_in[2];
    const float* w_s1v  = (const float*)d_in[3];
    const float* w_s2h  = (const float*)d_in[4];
    const float* w_s2v  = (const float*)d_in[5];
    const float* w_s3h  = (const float*)d_in[6];
    const float* w_s3v  = (const float*)d_in[7];
    const float* w_proj = (const float*)d_in[8];
    float* out = (float*)d_out;

    const size_t plane_bytes = (size_t)B_DIM * C_CH * HW * sizeof(float); // 128 MiB
    float* attn = (float*)d_ws;
    float* ssum = (float*)((char*)d_ws + plane_bytes);

    // K1: depthwise 5x5
    dim3 g1(W_DIM / 16, H_DIM / 16, B_DIM * C_CH);
    msca_dw5x5<<<g1, 256, 0, stream>>>(x, w_dw5, attn);

    // K2: fused separable multi-scale branches
    dim3 g2(W_DIM / 32, H_DIM / 32, B_DIM * C_CH);
    msca_sep<<<g2, 256, 0, stream>>>(attn, w_s1h, w_s1v, w_s2h, w_s2v,
                                     w_s3h, w_s3v, ssum);

    // K3: WMMA channel projection + sigmoid gating
    dim3 g3(HW / 64, B_DIM);
    msca_proj_gate<<<g3, 256, 0, stream>>>(ssum, w_proj, x, out);
}